// SelfAttention_78271484003205
// MI455X (gfx1250) — compile-verified
//
#include <hip/hip_runtime.h>
#include <math.h>
#include <stdint.h>

// ---------------- problem constants (from reference) ----------------
#define B_     2
#define N_     2048
#define DIM_   512
#define HEADS_ 8
#define DH_    64            // head dim
#define MF_    266           // NB_FEATURES
#define MP_    272           // padded to 17 tiles of 16
#define ROWS_  (B_ * N_)             // 4096 token rows
#define NTOK_  (B_ * HEADS_ * N_)    // 32768 (b,h,n) rows
#define CH_    64            // internal scan chunk (math is chunk-invariant)
#define EPS_   1e-4f
// LDS strides: 16B-aligned rows (for async b128 writes) + distinct-bank column maps
#define KSTR   276           // 276*4 B rows; bank = 20*row mod 64 -> 16 distinct
#define VSTR   68            // 68*4 B rows;  bank =  4*row mod 64 -> 16 distinct

typedef float v2f __attribute__((ext_vector_type(2)));
typedef float v8f __attribute__((ext_vector_type(8)));

__device__ __forceinline__ v8f vzero() {
  v8f z = {0.f, 0.f, 0.f, 0.f, 0.f, 0.f, 0.f, 0.f};
  return z;
}

// Native fp32 WMMA: D(16x16) = A(16x4) * B(4x16) + C, full fp32 precision.
__device__ __forceinline__ v8f wmma4(v2f a, v2f b, v8f c) {
  return __builtin_amdgcn_wmma_f32_16x16x4_f32(
      /*neg_a=*/false, a, /*neg_b=*/false, b,
      /*c_mod=*/(short)0, c, /*reuse_a=*/false, /*reuse_b=*/false);
}

// LDS aperture: generic pointer low 32 bits == LDS byte address (ISA 10.2)
__device__ __forceinline__ unsigned lds_off(const void* p) {
  return (unsigned)(uintptr_t)p;
}
// ASYNCcnt-tracked DMA: 16B per lane, global -> LDS, no VGPR round-trip
__device__ __forceinline__ void async_ld_b128(unsigned ldsaddr, const void* gaddr) {
  asm volatile("global_load_async_to_lds_b128 %0, %1, off"
               :: "v"(ldsaddr), "v"(gaddr) : "memory");
}
__device__ __forceinline__ void wait_async0() {
  asm volatile("s_wait_asynccnt 0" ::: "memory");
}

// order-preserving float<->uint for atomicMax on possibly-negative floats
__device__ __forceinline__ unsigned fenc(float x) {
  unsigned u = __float_as_uint(x);
  return (u >> 31) ? ~u : (u | 0x80000000u);
}
__device__ __forceinline__ float fdec(unsigned u) {
  return __uint_as_float((u >> 31) ? (u & 0x7FFFFFFFu) : ~u);
}

// ---------------- K1: qkv = x @ w_qkv, scattered into q/k/v [b,h,n,d] -----
__global__ __launch_bounds__(128)
void qkv_gemm(const float* __restrict__ x, const float* __restrict__ w,
              float* __restrict__ q, float* __restrict__ k, float* __restrict__ v) {
  const int lane = threadIdx.x & 31, wave = threadIdx.x >> 5;
  const int hi = lane >> 4, lo = lane & 15;
  const int mrow0 = blockIdx.x * 16;
  const int col0  = (blockIdx.y * 4 + wave) * 16;
  const int arow  = mrow0 + lo;
  v8f acc = vzero();
  for (int kk = 0; kk < DIM_; kk += 4) {
    const int ka = kk + 2 * hi;
    v2f a, b;
    a.x = x[(size_t)arow * DIM_ + ka];
    a.y = x[(size_t)arow * DIM_ + ka + 1];
    b.x = w[(size_t)ka * (3 * DIM_) + col0 + lo];
    b.y = w[(size_t)(ka + 1) * (3 * DIM_) + col0 + lo];
    acc = wmma4(a, b, acc);
  }
  const int col = col0 + lo;
  const int which = col / DIM_;
  const int jj = col % DIM_;
  const int h = jj / DH_, dd = jj % DH_;
  float* dst = (which == 0) ? q : (which == 1) ? k : v;
  for (int t = 0; t < 8; ++t) {
    const int r = mrow0 + t + 8 * hi;
    const int bb = r / N_, nn = r % N_;
    dst[((size_t)(bb * HEADS_ + h) * N_ + nn) * DH_ + dd] = acc[t];
  }
}

// ------- K2: dash[NTOK, MP] = (d^-0.25 * data) @ proj^T (zero-padded cols) --
__global__ __launch_bounds__(256)
void feat_gemm(const float* __restrict__ data, const float* __restrict__ proj,
               float* __restrict__ dash) {
  const int lane = threadIdx.x & 31, wave = threadIdx.x >> 5;
  const int hi = lane >> 4, lo = lane & 15;
  const int rowtile = blockIdx.x * 8 + wave;
  const int arow = rowtile * 16 + lo;
  const float nrm = 0.35355339059327373f;  // 64^-0.25
  for (int nt = 0; nt < MP_ / 16; ++nt) {
    const int col = nt * 16 + lo;
    v8f acc = vzero();
    for (int kk = 0; kk < DH_; kk += 4) {
      const int ka = kk + 2 * hi;
      v2f a, b;
      a.x = data[(size_t)arow * DH_ + ka];
      a.y = data[(size_t)arow * DH_ + ka + 1];
      b.x = (col < MF_) ? nrm * proj[(size_t)col * DH_ + ka]     : 0.f;
      b.y = (col < MF_) ? nrm * proj[(size_t)col * DH_ + ka + 1] : 0.f;
      acc = wmma4(a, b, acc);
    }
    for (int t = 0; t < 8; ++t)
      dash[(size_t)(rowtile * 16 + t + 8 * hi) * MP_ + col] = acc[t];
  }
}

// --- K3q: qp = ratio*(exp(dash - diag - rowmax)+EPS), in place, pad->0 -----
__global__ __launch_bounds__(256)
void q_exp(const float* __restrict__ qdata, float* __restrict__ dash) {
  const int lane = threadIdx.x & 31, wave = threadIdx.x >> 5;
  const int row = blockIdx.x * 8 + wave;
  float ss = 0.f;
  for (int d = lane; d < DH_; d += 32) {
    const float t = qdata[(size_t)row * DH_ + d];
    ss += t * t;
  }
  for (int o = 16; o; o >>= 1) ss += __shfl_xor(ss, o, 32);
  const float diag = 0.0625f * ss;  // 0.5 * 64^-0.5 * sumsq
  float mx = -3.4e38f;
  for (int m = lane; m < MF_; m += 32) mx = fmaxf(mx, dash[(size_t)row * MP_ + m]);
  for (int o = 16; o; o >>= 1) mx = fmaxf(mx, __shfl_xor(mx, o, 32));
  const float ratio = 1.0f / sqrtf((float)MF_);
  for (int m = lane; m < MP_; m += 32) {
    float val = 0.f;
    if (m < MF_) val = ratio * (expf(dash[(size_t)row * MP_ + m] - diag - mx) + EPS_);
    dash[(size_t)row * MP_ + m] = val;
  }
}

__global__ void init_gmax(unsigned* gmax) {
  if (threadIdx.x == 0) *gmax = fenc(-3.4e38f);
}

// --- K3k pass1: per-row diag, global max of kdash via atomicMax ------------
__global__ __launch_bounds__(256)
void k_stat(const float* __restrict__ kdata, const float* __restrict__ dash,
            float* __restrict__ diagk, unsigned* __restrict__ gmax) {
  const int lane = threadIdx.x & 31, wave = threadIdx.x >> 5;
  const int row = blockIdx.x * 8 + wave;
  float ss = 0.f;
  for (int d = lane; d < DH_; d += 32) {
    const float t = kdata[(size_t)row * DH_ + d];
    ss += t * t;
  }
  for (int o = 16; o; o >>= 1) ss += __shfl_xor(ss, o, 32);
  if (lane == 0) diagk[row] = 0.0625f * ss;
  float mx = -3.4e38f;
  for (int m = lane; m < MF_; m += 32) mx = fmaxf(mx, dash[(size_t)row * MP_ + m]);
  for (int o = 16; o; o >>= 1) mx = fmaxf(mx, __shfl_xor(mx, o, 32));
  if (lane == 0) atomicMax(gmax, fenc(mx));
}

// --- K3k pass2: kp = ratio*(exp(dash - diag - gmax)+EPS), in place ---------
__global__ __launch_bounds__(256)
void k_exp(float* __restrict__ dash, const float* __restrict__ diagk,
           const unsigned* __restrict__ gmax) {
  const int lane = threadIdx.x & 31, wave = threadIdx.x >> 5;
  const int row = blockIdx.x * 8 + wave;
  const float stab = fdec(*gmax);
  const float diag = diagk[row];
  const float ratio = 1.0f / sqrtf((float)MF_);
  for (int m = lane; m < MP_; m += 32) {
    float val = 0.f;
    if (m < MF_) val = ratio * (expf(dash[(size_t)row * MP_ + m] - diag - stab) + EPS_);
    dash[(size_t)row * MP_ + m] = val;
  }
}

// --- K5: causal linear attention scan, one workgroup per (b,h) -------------
// out = Q'@ctx_prev + tril(Q'@K^T)@V ;  ctx += K^T@V ;  ksum += colsum(K)
__global__ __launch_bounds__(256)
void scan_attn(const float* __restrict__ qp, const float* __restrict__ kp,
               const float* __restrict__ v, float* __restrict__ attn) {
  extern __shared__ float sm[];
  float* s_kp  = sm;                       // [CH_][KSTR]  K-features chunk
  float* s_qp  = s_kp  + CH_ * KSTR;       // [CH_][KSTR]  qp chunk -> q' in place
  float* s_v   = s_qp  + CH_ * KSTR;       // [CH_][VSTR]  V chunk
  float* s_S   = s_v   + CH_ * VSTR;       // [CH_][VSTR]  masked scores
  float* s_ctx = s_S   + CH_ * VSTR;       // [MP_][VSTR]  running K^T V state
  float* s_ks  = s_ctx + MP_ * VSTR;       // [MP_]        running ksum
  const int tid = threadIdx.x, lane = tid & 31, wave = tid >> 5;
  const int hi = lane >> 4, lo = lane & 15;
  const size_t R0 = (size_t)blockIdx.x * N_;

  for (int i = tid; i < MP_ * VSTR; i += 256) s_ctx[i] = 0.f;
  for (int m = tid; m < MP_; m += 256) s_ks[m] = (m < MF_) ? 0.f : 1.f; // pad->1 avoids 0/0
  __syncthreads();

  for (int cc = 0; cc < N_ / CH_; ++cc) {
    const size_t rb = R0 + (size_t)cc * CH_;
    // --- async DMA staging (ASYNCcnt path): kp, qp, v chunks -> LDS --------
    for (int i = tid; i < CH_ * (MP_ / 4); i += 256) {  // 17 iter/thread
      const int c = i / (MP_ / 4), g = (i % (MP_ / 4)) * 4;
      async_ld_b128(lds_off(s_kp + c * KSTR + g), kp + (rb + c) * MP_ + g);
      async_ld_b128(lds_off(s_qp + c * KSTR + g), qp + (rb + c) * MP_ + g);
    }
    for (int i = tid; i < CH_ * (DH_ / 4); i += 256) {  // 4 iter/thread
      const int c = i / (DH_ / 4), g = (i % (DH_ / 4)) * 4;
      async_ld_b128(lds_off(s_v + c * VSTR + g), v + (rb + c) * DH_ + g);
    }
    wait_async0();
    __syncthreads();
    // --- per-column inclusive cumsum of K; q' = qp/(ksum_prev+cumsum) ------
    for (int m = tid; m < MP_; m += 256) {
      float acc = 0.f;
      const float kprev = s_ks[m];
      for (int c = 0; c < CH_; ++c) {
        acc += s_kp[c * KSTR + m];
        s_qp[c * KSTR + m] = s_qp[c * KSTR + m] / (kprev + acc);
      }
      s_ks[m] = kprev + acc;
    }
    __syncthreads();
    // speculative prefetch of next chunk (global_prefetch_b8), overlaps WMMA
    if (cc + 1 < N_ / CH_) {
      const size_t rbn = rb + CH_;
      for (int i = tid; i < CH_; i += 256) {
        __builtin_prefetch(kp + (rbn + i) * MP_, 0, 1);
        __builtin_prefetch(qp + (rbn + i) * MP_, 0, 1);
      }
    }
    // --- out1 = Q' @ ctx_prev (4x4 tiles, 2 per wave), accumulate in regs --
    v8f out[2];
    for (int j = 0; j < 2; ++j) {
      const int tt = wave * 2 + j, mt = tt >> 2, nt = tt & 3;
      v8f acc = vzero();
      for (int kk = 0; kk < MP_; kk += 4) {
        const int ka = kk + 2 * hi;
        v2f a, b;
        a.x = s_qp[(mt * 16 + lo) * KSTR + ka];
        a.y = s_qp[(mt * 16 + lo) * KSTR + ka + 1];
        b.x = s_ctx[ka * VSTR + nt * 16 + lo];
        b.y = s_ctx[(ka + 1) * VSTR + nt * 16 + lo];
        acc = wmma4(a, b, acc);
      }
      out[j] = acc;
    }
    // --- S = tril(Q' @ K^T) (4x4 tiles, 2 per wave; above-diag tiles = 0) --
    for (int j = 0; j < 2; ++j) {
      const int tt = wave * 2 + j, mt = tt >> 2, it = tt & 3;
      v8f acc = vzero();
      if (it <= mt) {
        for (int kk = 0; kk < MP_; kk += 4) {
          const int ka = kk + 2 * hi;
          v2f a, b;
          a.x = s_qp[(mt * 16 + lo) * KSTR + ka];
          a.y = s_qp[(mt * 16 + lo) * KSTR + ka + 1];
          b.x = s_kp[(it * 16 + lo) * KSTR + ka];
          b.y = s_kp[(it * 16 + lo) * KSTR + ka + 1];
          acc = wmma4(a, b, acc);
        }
      }
      for (int t = 0; t < 8; ++t) {
        const int c = mt * 16 + t + 8 * hi;  // query row in chunk
        const int i = it * 16 + lo;          // key row in chunk
        s_S[c * VSTR + i] = (i <= c) ? acc[t] : 0.f;
      }
    }
    __syncthreads();
    // --- out2 = S @ V, fused add with out1, store attn[b,h,n,d] ------------
    for (int j = 0; j < 2; ++j) {
      const int tt = wave * 2 + j, mt = tt >> 2, nt = tt & 3;
      v8f acc = out[j];
      for (int kk = 0; kk < CH_; kk += 4) {
        const int ka = kk + 2 * hi;
        v2f a, b;
        a.x = s_S[(mt * 16 + lo) * VSTR + ka];
        a.y = s_S[(mt * 16 + lo) * VSTR + ka + 1];
        b.x = s_v[ka * VSTR + nt * 16 + lo];
        b.y = s_v[(ka + 1) * VSTR + nt * 16 + lo];
        acc = wmma4(a, b, acc);
      }
      for (int t = 0; t < 8; ++t)
        attn[(rb + mt * 16 + t + 8 * hi) * DH_ + nt * 16 + lo] = acc[t];
    }
    // --- ctx += K^T @ V (17x4 tiles, strided across 8 waves) ---------------
    for (int tt = wave; tt < (MP_ / 16) * 4; tt += 8) {
      const int mt = tt >> 2, nt = tt & 3;
      v8f acc;
      for (int t = 0; t < 8; ++t)
        acc[t] = s_ctx[(mt * 16 + t + 8 * hi) * VSTR + nt * 16 + lo];
      for (int kk = 0; kk < CH_; kk += 4) {
        const int ka = kk + 2 * hi;
        v2f a, b;
        a.x = s_kp[ka * KSTR + mt * 16 + lo];        // A[M=m, K=c] = K^T
        a.y = s_kp[(ka + 1) * KSTR + mt * 16 + lo];
        b.x = s_v[ka * VSTR + nt * 16 + lo];
        b.y = s_v[(ka + 1) * VSTR + nt * 16 + lo];
        acc = wmma4(a, b, acc);
      }
      for (int t = 0; t < 8; ++t)
        s_ctx[(mt * 16 + t + 8 * hi) * VSTR + nt * 16 + lo] = acc[t];
    }
    __syncthreads();
  }
}

// --- K6: out = attn(reshaped b,n,h*d) @ w_out + b_out ----------------------
__global__ __launch_bounds__(128)
void out_gemm(const float* __restrict__ attn, const float* __restrict__ w,
              const float* __restrict__ bias, float* __restrict__ out) {
  const int lane = threadIdx.x & 31, wave = threadIdx.x >> 5;
  const int hi = lane >> 4, lo = lane & 15;
  const int mrow0 = blockIdx.x * 16;
  const int col0  = (blockIdx.y * 4 + wave) * 16;
  const int r = mrow0 + lo;
  const int bb = r / N_, nn = r % N_;
  v8f acc = vzero();
  for (int kk = 0; kk < DIM_; kk += 4) {
    const int ka = kk + 2 * hi;                  // even, stays within one head
    const int h = ka / DH_, dd = ka % DH_;
    const size_t base = ((size_t)(bb * HEADS_ + h) * N_ + nn) * DH_;
    v2f a, b;
    a.x = attn[base + dd];
    a.y = attn[base + dd + 1];
    b.x = w[(size_t)ka * DIM_ + col0 + lo];
    b.y = w[(size_t)(ka + 1) * DIM_ + col0 + lo];
    acc = wmma4(a, b, acc);
  }
  const int col = col0 + lo;
  const float bo = bias[col];
  for (int t = 0; t < 8; ++t)
    out[(size_t)(mrow0 + t + 8 * hi) * DIM_ + col] = acc[t] + bo;
}

// ---------------------------------------------------------------------------
extern "C" void kernel_launch(void* const* d_in, const int* in_sizes, int n_in,
                              void* d_out, int out_size, void* d_ws, size_t ws_size,
                              hipStream_t stream) {
  const float* x     = (const float*)d_in[0];
  const float* w_qkv = (const float*)d_in[1];
  const float* w_out = (const float*)d_in[2];
  const float* b_out = (const float*)d_in[3];
  const float* proj  = (const float*)d_in[4];
  float* out = (float*)d_out;

  float* ws = (float*)d_ws;
  size_t off = 0;
  float* q     = ws + off; off += (size_t)NTOK_ * DH_;
  float* k     = ws + off; off += (size_t)NTOK_ * DH_;
  float* v     = ws + off; off += (size_t)NTOK_ * DH_;
  float* attn  = ws + off; off += (size_t)NTOK_ * DH_;
  float* qdash = ws + off; off += (size_t)NTOK_ * MP_;
  float* kdash = ws + off; off += (size_t)NTOK_ * MP_;
  float* diagk = ws + off; off += (size_t)NTOK_;
  unsigned* gmax = (unsigned*)(ws + off);

  const size_t smem =
      (size_t)(CH_ * KSTR * 2 + CH_ * VSTR * 2 + MP_ * VSTR + MP_) * sizeof(float);
  (void)hipFuncSetAttribute((const void*)scan_attn,
                            hipFuncAttributeMaxDynamicSharedMemorySize, (int)smem);

  qkv_gemm<<<dim3(ROWS_ / 16, (3 * DIM_) / 64), 128, 0, stream>>>(x, w_qkv, q, k, v);
  feat_gemm<<<dim3(NTOK_ / 128), 256, 0, stream>>>(q, proj, qdash);
  feat_gemm<<<dim3(NTOK_ / 128), 256, 0, stream>>>(k, proj, kdash);
  q_exp<<<dim3(NTOK_ / 8), 256, 0, stream>>>(q, qdash);
  init_gmax<<<1, 32, 0, stream>>>(gmax);
  k_stat<<<dim3(NTOK_ / 8), 256, 0, stream>>>(k, kdash, diagk, gmax);
  k_exp<<<dim3(NTOK_ / 8), 256, 0, stream>>>(kdash, diagk, gmax);
  scan_attn<<<dim3(B_ * HEADS_), 256, smem, stream>>>(qdash, kdash, v, attn);
  out_gemm<<<dim3(ROWS_ / 16, DIM_ / 64), 128, 0, stream>>>(attn, w_out, b_out, out);
}